// FrankWolfePolicyImprovement_73203422593452
// MI455X (gfx1250) — compile-verified
//
#include <hip/hip_runtime.h>
#include <math.h>

// ---------------- problem constants (from reference) ----------------
#define N_ACT_C   64
#define M_CON_C   32
#define SOFT_C    4
#define HARD_C    28           // M_CON - SOFT
#define HID_C     256
#define ITERS_C   80
static __device__ constexpr float ALPHA_C  = 0.1f;
static __device__ constexpr float SLACKP_C = 1.0f;
static __device__ constexpr float RHO_C    = 1.0f;
static __device__ constexpr float SIGMA_C  = 1e-6f;
static __device__ constexpr float EPS_C    = 1e-4f;

#define T    128               // threads per block = 4 wave32
#define LDA  68                // LDS stride for A rows (32 x 64 payload)
#define LDM  80                // LDS stride for M/L/J (68 rows, padded to 5 WMMA tiles)
#define LDW  72                // LDS stride for W = M^-1 (and Gram scratch)

typedef float v2f __attribute__((ext_vector_type(2)));
typedef float v8f __attribute__((ext_vector_type(8)));

// ---------------------------------------------------------------
// Gram product D[lim x lim] = S^T S, S is K x (16*nT) in LDS, via
// V_WMMA_F32_16X16X4_F32.  K must be a multiple of 4.  Wave-uniform
// tile loop => EXEC stays all-ones at every WMMA (ISA requirement).
// A-operand(m,k) = S[k][ti*16+m], B-operand(k,n) = S[k][tj*16+n].
// ---------------------------------------------------------------
__device__ __forceinline__ void gram_wmma(const float* S, int ldS, int K,
                                          float* D, int ldD, int nT, int lim,
                                          int tid)
{
#if __has_builtin(__builtin_amdgcn_wmma_f32_16x16x4_f32)
  const int lane = tid & 31;
  const int wave = tid >> 5;
  const int half = lane >> 4;       // K-half selector (ISA 16x4 f32 A layout)
  const int l15  = lane & 15;
  const int nTiles = nT * nT;
  for (int t = wave; t < nTiles; t += T / 32) {   // wave-uniform trip count
    const int ti = t / nT;
    const int tj = t % nT;
    const float* pa = S + ti * 16 + l15;
    const float* pb = S + tj * 16 + l15;
    v8f acc = {};
    for (int kb = 0; kb < K; kb += 4) {
      const int k0 = (kb + 2 * half) * ldS;       // VGPR0: K = 0/2, VGPR1: K = 1/3
      v2f a, b;
      a[0] = pa[k0];        a[1] = pa[k0 + ldS];
      b[0] = pb[k0];        b[1] = pb[k0 + ldS];
      acc = __builtin_amdgcn_wmma_f32_16x16x4_f32(false, a, false, b,
                                                  (short)0, acc, false, false);
    }
#pragma unroll
    for (int v = 0; v < 8; ++v) {                 // D: lane l15 = col, v+8*half = row
      const int row = ti * 16 + v + 8 * half;
      const int col = tj * 16 + l15;
      if (row < lim && col < lim) D[row * ldD + col] = acc[v];
    }
  }
#else
  // VALU fallback (should not trigger on gfx1250)
  for (int idx = tid; idx < lim * lim; idx += T) {
    int i = idx / lim, j = idx % lim;
    float s = 0.0f;
    for (int k = 0; k < K; ++k) s += S[k * ldS + i] * S[k * ldS + j];
    D[i * ldD + j] = s;
  }
#endif
}

// ---------------------------------------------------------------
// In-place Cholesky (lower) of dxd SPD matrix in sM (stride LDM),
// then in-place triangular inversion L -> J = L^-1; rows are padded
// with zeros out to LDM columns so J can feed gram_wmma directly.
// ---------------------------------------------------------------
__device__ void chol_invert(float* Mtx, int d, int tid, float* sScal)
{
  for (int j = 0; j < d; ++j) {
    if (tid == 0) {
      float lj = sqrtf(Mtx[j * LDM + j]);
      Mtx[j * LDM + j] = lj;
      sScal[0] = 1.0f / lj;
    }
    __syncthreads();
    const float rd = sScal[0];
    for (int i = j + 1 + tid; i < d; i += T) Mtx[i * LDM + j] *= rd;
    __syncthreads();
    for (int i = j + 1 + tid; i < d; i += T) {       // rank-1, lower triangle only
      const float lij = Mtx[i * LDM + j];
      for (int k = j + 1; k <= i; ++k)
        Mtx[i * LDM + k] -= lij * Mtx[k * LDM + j];
    }
    __syncthreads();
  }
  // J[i][j] = -(sum_{k=j..i-1} L[i][k] J[k][j]) / L[i][i]; row i of L is
  // consumed then overwritten by row i of J (valid in-place ordering).
  for (int i = 0; i < d; ++i) {
    const float rLii = 1.0f / Mtx[i * LDM + i];
    float val = 0.0f;
    if (tid < i) {
      float s = 0.0f;
      for (int k = tid; k < i; ++k)
        s += Mtx[i * LDM + k] * Mtx[k * LDM + tid];
      val = -s * rLii;
    } else if (tid == i) {
      val = rLii;
    }
    __syncthreads();
    if (tid < LDM) Mtx[i * LDM + tid] = (tid <= i) ? val : 0.0f;  // zero-pad row
    __syncthreads();
  }
}

// ---------------------------------------------------------------
// Fused Frank-Wolfe policy improvement: one workgroup per problem.
// ---------------------------------------------------------------
__global__ __launch_bounds__(T)
void fw_policy_kernel(const float* __restrict__ xraw_g,
                      const float* __restrict__ A_g,
                      const float* __restrict__ b_g,
                      const float* __restrict__ W1_g,
                      const float* __restrict__ w2_g,
                      float* __restrict__ out_g)
{
  __shared__ float sA[M_CON_C * LDA];   // A rows (later An)
  __shared__ float sM[LDA * LDM];       // M -> L -> J
  __shared__ float sW[LDA * LDW];       // Gram scratch / W = M^-1
  __shared__ float sXraw[N_ACT_C];
  __shared__ float sB[M_CON_C];         // b (later bn)
  __shared__ float sXf[N_ACT_C];        // anchor solution
  __shared__ float sG[N_ACT_C];         // MLP gradient
  __shared__ float sH[HID_C];
  __shared__ float sx[LDA];
  __shared__ float sz[100];
  __shared__ float sy[100];
  __shared__ float st_[100];
  __shared__ float srhs[LDA];
  __shared__ float sScal[1];

  const int p   = blockIdx.x;
  const int tid = threadIdx.x;
  const float* Ap = A_g + (size_t)p * (M_CON_C * N_ACT_C);

  for (int idx = tid; idx < M_CON_C * N_ACT_C; idx += T) {
    const int r = idx >> 6, i = idx & 63;
    sA[r * LDA + i] = Ap[idx];
  }
  if (tid < N_ACT_C) sXraw[tid] = xraw_g[(size_t)p * N_ACT_C + tid];
  if (tid < M_CON_C) sB[tid]   = b_g[(size_t)p * M_CON_C + tid];
  __syncthreads();

  // ================= ANCHOR QP: d=68 (x,s), k=100 =================
  // C = [I 0; Ah 0; As -I; 0 I]  =>  M = [[ (1+s+r)I + r A^T A , -r As^T ],
  //                                       [ -r As , (2P+s+2r) I ]]
  gram_wmma(sA, LDA, M_CON_C, sW, LDW, 4, N_ACT_C, tid);   // G = A^T A
  __syncthreads();
  for (int idx = tid; idx < 68 * LDM; idx += T) {
    const int i = idx / LDM, j = idx % LDM;
    float v = 0.0f;
    if (i < 64 && j < 64) {
      v = RHO_C * sW[i * LDW + j];
      if (i == j) v += 1.0f + SIGMA_C + RHO_C;
    } else if (i < 64 && j >= 64 && j < 68) {
      v = -RHO_C * sA[(HARD_C + (j - 64)) * LDA + i];
    } else if (i >= 64 && i < 68 && j < 64) {
      v = -RHO_C * sA[(HARD_C + (i - 64)) * LDA + j];
    } else if (i >= 64 && i < 68 && j >= 64 && j < 68) {
      v = (i == j) ? (2.0f * SLACKP_C + SIGMA_C + 2.0f * RHO_C) : 0.0f;
    }
    sM[idx] = v;                                           // cols 68..79 zero-padded
  }
  __syncthreads();
  chol_invert(sM, 68, tid, sScal);                         // sM := J = L^-1
  gram_wmma(sM, LDM, 68, sW, LDW, 5, 68, tid);             // W = J^T J = M^-1
  for (int i = tid; i < 100; i += T) { sz[i] = 0.0f; sy[i] = 0.0f; }
  for (int i = tid; i < 68;  i += T) sx[i] = 0.0f;
  __syncthreads();

  for (int it = 0; it < ITERS_C; ++it) {
    for (int k = tid; k < 100; k += T) st_[k] = sz[k] - sy[k];   // rho*z - y, rho=1
    __syncthreads();
    if (tid < 64) {                                        // rhs_x = s*x + xraw + t_x + A^T t_A
      float a = SIGMA_C * sx[tid] + sXraw[tid] + st_[tid];
      for (int r = 0; r < M_CON_C; ++r) a += sA[r * LDA + tid] * st_[64 + r];
      srhs[tid] = a;
    } else if (tid < 68) {                                 // rhs_s = s*s - t_soft + t_sbnd
      const int t4 = tid - 64;
      srhs[tid] = SIGMA_C * sx[tid] - st_[92 + t4] + st_[96 + t4];
    }
    __syncthreads();
    if (tid < 68) {                                        // x = W * rhs (dense, parallel)
      float a = 0.0f;
      for (int j = 0; j < 68; ++j) a += sW[tid * LDW + j] * srhs[j];
      sx[tid] = a;
    }
    __syncthreads();
    if (tid < 100) {                                       // z = clip(Cx + y), y += Cx - z
      float c;
      if (tid < 64)      c = sx[tid];
      else if (tid < 96) {
        const int r = tid - 64;
        float a = 0.0f;
        for (int i = 0; i < 64; ++i) a += sA[r * LDA + i] * sx[i];
        if (r >= HARD_C) a -= sx[64 + (r - HARD_C)];       // soft rows: As x - s
        c = a;
      } else             c = sx[64 + (tid - 96)];
      const float v  = c + sy[tid];
      float zn;
      if (tid < 64)      zn = fmaxf(v, 0.0f);              // x >= LOWER
      else if (tid < 96) zn = fminf(v, sB[tid - 64]);      // A x <= b
      else               zn = fmaxf(v, 0.0f);              // s >= 0
      sy[tid] = v - zn;
      sz[tid] = zn;
    }
    __syncthreads();
  }
  if (tid < 64) sXf[tid] = sx[tid];
  __syncthreads();

  // ================= MLP gradient + row normalization =================
  for (int c = tid; c < HID_C; c += T) {
    float a = 0.0f;
    for (int i = 0; i < 64; ++i) a += sXf[i] * W1_g[i * HID_C + c];
    const float h = tanhf(a);
    sH[c] = (1.0f - h * h) * w2_g[c];
  }
  __syncthreads();
  if (tid < 64) {                                          // g = u @ W1^T
    float a = 0.0f;
    const float* w1row = W1_g + tid * HID_C;
    for (int c2 = 0; c2 < HID_C; ++c2) a += sH[c2] * w1row[c2];
    sG[tid] = a;
  } else if (tid < 96) {                                   // An = A/||A||, bn = b/||A||
    const int r = tid - 64;
    float n2 = 0.0f;
    for (int i = 0; i < 64; ++i) { const float a = sA[r * LDA + i]; n2 += a * a; }
    const float rinv = 1.0f / fmaxf(sqrtf(n2), 1e-12f);
    for (int i = 0; i < 64; ++i) sA[r * LDA + i] *= rinv;
    sB[r] *= rinv;
  }
  __syncthreads();

  // ================= LMO QP: d=64, k=96, C=[I; An] =================
  gram_wmma(sA, LDA, M_CON_C, sW, LDW, 4, N_ACT_C, tid);   // G = An^T An
  __syncthreads();
  for (int idx = tid; idx < 64 * LDM; idx += T) {
    const int i = idx / LDM, j = idx % LDM;
    float v = 0.0f;
    if (j < 64) {
      v = RHO_C * sW[i * LDW + j];
      if (i == j) v += EPS_C + SIGMA_C + RHO_C;
    }
    sM[idx] = v;
  }
  __syncthreads();
  chol_invert(sM, 64, tid, sScal);
  gram_wmma(sM, LDM, 64, sW, LDW, 4, 64, tid);             // W = M^-1
  for (int i = tid; i < 96; i += T) { sz[i] = 0.0f; sy[i] = 0.0f; }
  if (tid < 64) sx[tid] = 0.0f;
  __syncthreads();

  for (int it = 0; it < ITERS_C; ++it) {
    for (int k = tid; k < 96; k += T) st_[k] = sz[k] - sy[k];
    __syncthreads();
    if (tid < 64) {                                        // q = -g  =>  -q = +g
      float a = SIGMA_C * sx[tid] + sG[tid] + st_[tid];
      for (int r = 0; r < M_CON_C; ++r) a += sA[r * LDA + tid] * st_[64 + r];
      srhs[tid] = a;
    }
    __syncthreads();
    if (tid < 64) {
      float a = 0.0f;
      for (int j = 0; j < 64; ++j) a += sW[tid * LDW + j] * srhs[j];
      sx[tid] = a;
    }
    __syncthreads();
    if (tid < 96) {
      float c;
      if (tid < 64) c = sx[tid];
      else {
        const int r = tid - 64;
        float a = 0.0f;
        for (int i = 0; i < 64; ++i) a += sA[r * LDA + i] * sx[i];
        c = a;
      }
      const float v  = c + sy[tid];
      const float zn = (tid < 64) ? fmaxf(v, 0.0f) : fminf(v, sB[tid - 64]);
      sy[tid] = v - zn;
      sz[tid] = zn;
    }
    __syncthreads();
  }

  // ================= Frank-Wolfe blend =================
  if (tid < 64)
    out_g[(size_t)p * N_ACT_C + tid] =
        (1.0f - ALPHA_C) * sXf[tid] + ALPHA_C * sx[tid];
}

extern "C" void kernel_launch(void* const* d_in, const int* in_sizes, int n_in,
                              void* d_out, int out_size, void* d_ws, size_t ws_size,
                              hipStream_t stream)
{
  (void)n_in; (void)out_size; (void)d_ws; (void)ws_size;
  const float* x_raw = (const float*)d_in[0];
  const float* A     = (const float*)d_in[1];
  const float* b     = (const float*)d_in[2];
  const float* W1    = (const float*)d_in[3];
  const float* w2    = (const float*)d_in[4];
  float* out = (float*)d_out;
  const int nprob = in_sizes[0] / N_ACT_C;   // B*S = 4096 independent problems
  fw_policy_kernel<<<nprob, T, 0, stream>>>(x_raw, A, b, W1, w2, out);
}